// EGCN_86449101734541
// MI455X (gfx1250) — compile-verified
//
#include <hip/hip_runtime.h>
#include <stdint.h>

#define FD 128            // feature dim (32 lanes * 4 floats)
#define WPB 8             // waves per 256-thread block
#define NEG_SLOPE 0.01f
#define EPS_NORM 1e-12f

// ---------------------------------------------------------------- helpers
__device__ __forceinline__ float wave_sum32(float v) {
    v += __shfl_xor(v, 16, 32);
    v += __shfl_xor(v,  8, 32);
    v += __shfl_xor(v,  4, 32);
    v += __shfl_xor(v,  2, 32);
    v += __shfl_xor(v,  1, 32);
    return v;
}

// guaranteed hardware no-return f32 atomic (STOREcnt-tracked, L2 atomic unit)
__device__ __forceinline__ void atomAddF(float* p, float v) {
    asm volatile("global_atomic_add_f32 %0, %1, off"
                 :: "v"(p), "v"(v) : "memory");
}

// ---------------------------------------------------------------- kernels

// One wave32 per row: L2-normalize, write to xn (layer input) and out (residual x)
__global__ void egcn_normalize(const float* __restrict__ x,
                               float* __restrict__ xn,
                               float* __restrict__ out, int N) {
    int row  = (int)((blockIdx.x * blockDim.x + threadIdx.x) >> 5);
    int lane = threadIdx.x & 31;
    if (row >= N) return;
    const float4 v = ((const float4*)(x + (size_t)row * FD))[lane];
    float ss = v.x * v.x + v.y * v.y + v.z * v.z + v.w * v.w;
    ss = wave_sum32(ss);
    float inv = 1.0f / fmaxf(sqrtf(ss), EPS_NORM);
    float4 r;
    r.x = v.x * inv; r.y = v.y * inv; r.z = v.z * inv; r.w = v.w * inv;
    ((float4*)(xn  + (size_t)row * FD))[lane] = r;
    ((float4*)(out + (size_t)row * FD))[lane] = r;
}

__global__ void egcn_zero(float* __restrict__ p, size_t n4) {
    size_t i = (size_t)blockIdx.x * blockDim.x + threadIdx.x;
    if (i < n4) ((float4*)p)[i] = make_float4(0.f, 0.f, 0.f, 0.f);
}

// Edge-doubled in-degree: for edge e, cnt[row1[e]]++ and cnt[row0[e]]++
__global__ void egcn_degree(const int* __restrict__ ei, float* __restrict__ cnt, int E) {
    int e = blockIdx.x * blockDim.x + threadIdx.x;
    if (e < E) {
        int a = ei[e];         // row 0
        int b = ei[E + e];     // row 1
        atomAddF(&cnt[b], 1.0f);
        atomAddF(&cnt[a], 1.0f);
    }
}

__global__ void egcn_recip(float* __restrict__ cnt, int N) {
    int i = blockIdx.x * blockDim.x + threadIdx.x;
    if (i < N) cnt[i] = 1.0f / fmaxf(cnt[i], 1.0f);
}

// One wave per (doubled) edge, grid-stride. Gather x[src] row via async
// global->LDS copy, double-buffered; scatter with no-return f32 atomics.
// Doubled edge list: src = ei[e]; dst = ei[e<E ? e+E : e-E]; w = wv[e].
__global__ void egcn_scatter(const float* __restrict__ feat,
                             const int* __restrict__ ei,
                             const float* __restrict__ wv,
                             float* __restrict__ accum, int E) {
    __shared__ __align__(16) float stage[WPB][2][FD];
    const int lane = threadIdx.x & 31;
    const int wib  = threadIdx.x >> 5;
    const int gw   = blockIdx.x * WPB + wib;       // global wave id (uniform per wave)
    const int gW   = gridDim.x * WPB;
    const int total = 2 * E;
    if (gw >= total) return;

    float*   s0   = &stage[wib][0][lane * 4];
    float*   s1   = &stage[wib][1][lane * 4];
    uint32_t lds0 = (uint32_t)(uintptr_t)s0;       // low 32 bits = LDS byte offset
    uint32_t lds1 = (uint32_t)(uintptr_t)s1;

    // prologue: issue gather for first edge into buffer 0
    {
        int src = ei[gw];
        const float* g = feat + (size_t)src * FD + (size_t)(lane * 4);
        asm volatile("global_load_async_to_lds_b128 %0, %1, off"
                     :: "v"(lds0), "v"(g) : "memory");
    }

    int buf = 0;
    for (int e = gw; e < total; e += gW, buf ^= 1) {
        int nxt = e + gW;
        if (nxt < total) {                          // issue next gather into other buffer
            int srcn = ei[nxt];
            const float* gn = feat + (size_t)srcn * FD + (size_t)(lane * 4);
            asm volatile("global_load_async_to_lds_b128 %0, %1, off"
                         :: "v"(buf ? lds0 : lds1), "v"(gn) : "memory");
            asm volatile("s_wait_asynccnt 1" ::: "memory");   // current buffer ready
        } else {
            asm volatile("s_wait_asynccnt 0" ::: "memory");
        }

        float w   = wv[e];
        int   dst = ei[(e < E) ? (e + E) : (e - E)];
        const float4 v = *(const float4*)(buf ? s1 : s0);     // ds_read_b128

        float* base = accum + (size_t)dst * FD + (size_t)(lane * 4);
        atomAddF(base + 0, v.x * w);
        atomAddF(base + 1, v.y * w);
        atomAddF(base + 2, v.z * w);
        atomAddF(base + 3, v.w * w);
    }
    asm volatile("s_wait_asynccnt 0" ::: "memory");
}

// mean + leaky_relu; optionally store new features; accumulate into out
__global__ void egcn_finalize(const float* __restrict__ sum,
                              const float* __restrict__ rinv,
                              float* __restrict__ newfeat,
                              float* __restrict__ out, int N) {
    int t = blockIdx.x * blockDim.x + threadIdx.x;   // float4 index
    int total = N * (FD / 4);
    if (t >= total) return;
    int node = t / (FD / 4);
    float r = rinv[node];
    float4 s = ((const float4*)sum)[t];
    float4 y;
    float m;
    m = s.x * r; y.x = (m >= 0.f) ? m : NEG_SLOPE * m;
    m = s.y * r; y.y = (m >= 0.f) ? m : NEG_SLOPE * m;
    m = s.z * r; y.z = (m >= 0.f) ? m : NEG_SLOPE * m;
    m = s.w * r; y.w = (m >= 0.f) ? m : NEG_SLOPE * m;
    if (newfeat) ((float4*)newfeat)[t] = y;
    float4 o = ((const float4*)out)[t];
    o.x += y.x; o.y += y.y; o.z += y.z; o.w += y.w;
    ((float4*)out)[t] = o;
}

// ---------------------------------------------------------------- launch
extern "C" void kernel_launch(void* const* d_in, const int* in_sizes, int n_in,
                              void* d_out, int out_size, void* d_ws, size_t ws_size,
                              hipStream_t stream) {
    const float* x  = (const float*)d_in[0];   // [N, 128] f32
    const int*   ei = (const int*)d_in[1];     // [2, E] i32 (JAX default int)
    const float* wv = (const float*)d_in[2];   // [2E, 1] f32
    float* out = (float*)d_out;

    const int ND = in_sizes[0];
    const int N  = ND / FD;
    const int E  = in_sizes[1] / 2;

    float* bufA = (float*)d_ws;                // x (normalized), then x1
    float* bufB = bufA + (size_t)ND;           // scatter accumulator
    float* cnt  = bufB + (size_t)ND;           // degree -> reciprocal

    // 1. normalize -> bufA, out = x
    egcn_normalize<<<(N + WPB - 1) / WPB, 256, 0, stream>>>(x, bufA, out, N);

    // 2. zero bufB + cnt (contiguous)
    size_t z1 = ((size_t)ND + (size_t)N + 3) / 4;
    egcn_zero<<<(unsigned)((z1 + 255) / 256), 256, 0, stream>>>(bufB, z1);

    // 3-4. degree once (shared by both layers), then invert
    egcn_degree<<<(E + 255) / 256, 256, 0, stream>>>(ei, cnt, E);
    egcn_recip<<<(N + 255) / 256, 256, 0, stream>>>(cnt, N);

    int totWaves = 2 * E;
    int scatBlocks = (totWaves + WPB - 1) / WPB;
    if (scatBlocks > 4096) scatBlocks = 4096;

    // 5-6. layer 1: bufB = scatter(bufA); x1 -> bufA, out += x1
    egcn_scatter<<<scatBlocks, 256, 0, stream>>>(bufA, ei, wv, bufB, E);
    egcn_finalize<<<(ND / 4 + 255) / 256, 256, 0, stream>>>(bufB, cnt, bufA, out, N);

    // 7. re-zero accumulator
    size_t z2 = (size_t)ND / 4;
    egcn_zero<<<(unsigned)((z2 + 255) / 256), 256, 0, stream>>>(bufB, z2);

    // 8-9. layer 2: bufB = scatter(bufA=x1); out += x2
    egcn_scatter<<<scatBlocks, 256, 0, stream>>>(bufA, ei, wv, bufB, E);
    egcn_finalize<<<(ND / 4 + 255) / 256, 256, 0, stream>>>(bufB, cnt, nullptr, out, N);
}